// AttentionModule_85280870629694
// MI455X (gfx1250) — compile-verified
//
#include <hip/hip_runtime.h>

#define Bn 16
#define Cc 128
#define C4n 32
#define Hn 48
#define Wn 48
#define Nn (Hn * Wn)          // 2304
#define BN_EPS 1e-5f
#define NTILE (Nn / 16)       // 144
#define NGRP  (NTILE / 8)     // 18 n-tile groups (8 waves/block)
#define MGRP  (Nn / 128)      // 18 m chunks of 128
#define NPART (NTILE * MGRP)  // 2592 partials per batch

typedef __attribute__((ext_vector_type(2))) float v2f;
typedef __attribute__((ext_vector_type(8))) float v8f;

__device__ __forceinline__ v8f wmma4(v2f a, v2f b, v8f c) {
  // D(16x16 f32) = A(16x4 f32) * B(4x16 f32) + C
  return __builtin_amdgcn_wmma_f32_16x16x4_f32(
      false, a, false, b, (short)0, c, false, false);
}

// order-preserving float <-> uint map (exact, deterministic atomic max)
__device__ __forceinline__ unsigned flipf(float f) {
  unsigned u = __float_as_uint(f);
  return (u & 0x80000000u) ? ~u : (u | 0x80000000u);
}
__device__ __forceinline__ float unflipf(unsigned u) {
  u = (u & 0x80000000u) ? (u & 0x7FFFFFFFu) : ~u;
  return __uint_as_float(u);
}

// ======================= WMMA 1x1 conv kernels =======================
// All are channel GEMMs. BN scale is folded into the staged weights; bias
// (beta - mean*scale) is added in the epilogue, then ReLU.

// ---- K/Q conv: out[k][n] = relu(sum_c w[k][c] x[c][n] * s + bi), k<32 ----
// D[M=k][N=n]; A = scaled weights from LDS; B = x from global (coalesced).
// Output stored pair-interleaved on k: dst[(k>>1)*2N + 2n + (k&1)].
__global__ void conv_kq_wmma(const float* __restrict__ x_en,
                             const float* __restrict__ x_de,
                             const float* __restrict__ w,  // [C4][C]
                             const float* __restrict__ g, const float* __restrict__ be,
                             const float* __restrict__ mn, const float* __restrict__ vr,
                             float* __restrict__ Kp, float* __restrict__ Qp) {
  int ng = blockIdx.x, ot = blockIdx.y, z = blockIdx.z;  // z: 0..2B-1
  int b = z & (Bn - 1);
  const float* x = (z < Bn) ? x_en : x_de;
  float* dst = ((z < Bn) ? Kp : Qp) + (size_t)b * C4n * Nn;
  int o0 = ot * 16;
  __shared__ float Wa[16][132];  // bank = (132*o + c)%64 = (4o + c)%64 -> conflict-free
  for (int i = threadIdx.x; i < 16 * Cc; i += 256) {
    int oo = i >> 7, c = i & (Cc - 1);
    float s = g[o0 + oo] * rsqrtf(vr[o0 + oo] + BN_EPS);
    Wa[oo][c] = w[(o0 + oo) * Cc + c] * s;
  }
  __syncthreads();
  int wave = threadIdx.x >> 5, lane = threadIdx.x & 31;
  int lm = lane & 15, kh = (lane >> 4) << 1, rofs = (lane >> 4) << 3;
  int n0 = (ng * 8 + wave) * 16;
  const float* xb = x + (size_t)b * Cc * Nn;
  v8f acc = {};
#pragma unroll
  for (int c = 0; c < Cc; c += 4) {
    v2f a, bb;
    a.x = Wa[lm][c + kh];
    a.y = Wa[lm][c + kh + 1];
    bb.x = xb[(size_t)(c + kh) * Nn + n0 + lm];
    bb.y = xb[(size_t)(c + kh + 1) * Nn + n0 + lm];
    acc = wmma4(a, bb, acc);
  }
#pragma unroll
  for (int r = 0; r < 8; ++r) {
    int k = o0 + r + rofs;
    float s = g[k] * rsqrtf(vr[k] + BN_EPS);
    float v = acc[r] + (be[k] - mn[k] * s);
    dst[(size_t)(k >> 1) * (2 * Nn) + 2 * (n0 + lm) + (k & 1)] = v > 0.f ? v : 0.f;
  }
}

// ---- reduce conv: x_red[o][n] from x_cat (CIN=256), written to out[:,0:C,:] ----
__global__ void conv_red_wmma(const float* __restrict__ x_cat,
                              const float* __restrict__ w,  // [C][2C]
                              const float* __restrict__ g, const float* __restrict__ be,
                              const float* __restrict__ mn, const float* __restrict__ vr,
                              float* __restrict__ out) {
  int ng = blockIdx.x, ot = blockIdx.y, b = blockIdx.z;
  int o0 = ot * 16;
  __shared__ float Wa[16][260];  // bank = (4o + c)%64 -> conflict-free
  for (int i = threadIdx.x; i < 16 * 2 * Cc; i += 256) {
    int oo = i >> 8, c = i & (2 * Cc - 1);
    float s = g[o0 + oo] * rsqrtf(vr[o0 + oo] + BN_EPS);
    Wa[oo][c] = w[(o0 + oo) * 2 * Cc + c] * s;
  }
  __syncthreads();
  int wave = threadIdx.x >> 5, lane = threadIdx.x & 31;
  int lm = lane & 15, kh = (lane >> 4) << 1, rofs = (lane >> 4) << 3;
  int n0 = (ng * 8 + wave) * 16;
  const float* xb = x_cat + (size_t)b * 2 * Cc * Nn;
  v8f acc = {};
#pragma unroll
  for (int c = 0; c < 2 * Cc; c += 4) {
    v2f a, bb;
    a.x = Wa[lm][c + kh];
    a.y = Wa[lm][c + kh + 1];
    bb.x = xb[(size_t)(c + kh) * Nn + n0 + lm];
    bb.y = xb[(size_t)(c + kh + 1) * Nn + n0 + lm];
    acc = wmma4(a, bb, acc);
  }
#pragma unroll
  for (int r = 0; r < 8; ++r) {
    int o = o0 + r + rofs;
    float s = g[o] * rsqrtf(vr[o] + BN_EPS);
    float v = acc[r] + (be[o] - mn[o] * s);
    out[((size_t)b * 2 * Cc + o) * Nn + n0 + lm] = v > 0.f ? v : 0.f;  // coalesced in n
  }
}

// ---- V conv: Vp (pair-interleaved on n) from x_red (= d_out[:,0:C,:]) ----
// D[M=n][N=o]; A = x_red from global (coalesced); B = scaled W^T from LDS.
// Vp[(n>>1)*2C + 2c + (n&1)] so the feat GEMM gets contiguous (m, m+1) pairs.
__global__ void conv_v_wmma(const float* __restrict__ xred,  // d_out, [B][2C][N]
                            const float* __restrict__ w,     // [C][C]
                            const float* __restrict__ g, const float* __restrict__ be,
                            const float* __restrict__ mn, const float* __restrict__ vr,
                            float* __restrict__ Vp) {
  int ng = blockIdx.x, ot = blockIdx.y, b = blockIdx.z;
  int o0 = ot * 16;
  __shared__ float Wt[Cc][20];  // bank = (20c + o)%64; halves offset by 40 -> ok
  for (int i = threadIdx.x; i < Cc * 16; i += 256) {
    int oo = i >> 7, c = i & (Cc - 1);
    float s = g[o0 + oo] * rsqrtf(vr[o0 + oo] + BN_EPS);
    Wt[c][oo] = w[(o0 + oo) * Cc + c] * s;
  }
  __syncthreads();
  int wave = threadIdx.x >> 5, lane = threadIdx.x & 31;
  int lm = lane & 15, kh = (lane >> 4) << 1, rofs = (lane >> 4) << 3;
  int n0 = (ng * 8 + wave) * 16;
  const float* xb = xred + (size_t)b * 2 * Cc * Nn;
  int o = o0 + lm;
  float s = g[o] * rsqrtf(vr[o] + BN_EPS);
  float bias = be[o] - mn[o] * s;
  v8f acc = {};
#pragma unroll
  for (int c = 0; c < Cc; c += 4) {
    v2f a, bb;
    a.x = xb[(size_t)(c + kh) * Nn + n0 + lm];
    a.y = xb[(size_t)(c + kh + 1) * Nn + n0 + lm];
    bb.x = Wt[c + kh][lm];
    bb.y = Wt[c + kh + 1][lm];
    acc = wmma4(a, bb, acc);
  }
  float* Vb = Vp + (size_t)b * Nn * Cc;
#pragma unroll
  for (int r = 0; r < 8; ++r) {
    int n = n0 + r + rofs;
    float v = acc[r] + bias;
    Vb[(size_t)(n >> 1) * (2 * Cc) + 2 * o + (n & 1)] = v > 0.f ? v : 0.f;
  }
}

// ======================= KQ tile (pair-interleaved K/Q, b64 loads) =======================
__device__ __forceinline__ v8f kq_tile(const float* __restrict__ Kb,
                                       const float* __restrict__ Qb,
                                       int n0, int m0, int lm, int khh) {
  v8f acc = {};
#pragma unroll
  for (int j = 0; j < 8; ++j) {
    v2f a = *(const v2f*)(Kb + (size_t)(2 * j + khh) * (2 * Nn) + 2 * (n0 + lm));
    v2f bb = *(const v2f*)(Qb + (size_t)(2 * j + khh) * (2 * Nn) + 2 * (m0 + lm));
    acc = wmma4(a, bb, acc);
  }
  return acc;
}

// ---------------- pass 1: per-batch global max of KQ ----------------
__global__ void kq_max_kernel(const float* __restrict__ Kp,
                              const float* __restrict__ Qp,
                              unsigned* __restrict__ bmax) {
  int mg = blockIdx.x, nt = blockIdx.y, b = blockIdx.z;
  int wave = threadIdx.x >> 5, lane = threadIdx.x & 31;
  int lm = lane & 15, khh = lane >> 4;
  const float* Kb = Kp + (size_t)b * C4n * Nn;
  const float* Qb = Qp + (size_t)b * C4n * Nn;
  v8f t = kq_tile(Kb, Qb, nt * 16, mg * 128 + wave * 16, lm, khh);
  float mx = t[0];
#pragma unroll
  for (int r = 1; r < 8; ++r) mx = fmaxf(mx, t[r]);
  __shared__ float red[256];
  red[threadIdx.x] = mx;
  __syncthreads();
  for (int s = 128; s > 0; s >>= 1) {
    if (threadIdx.x < s) red[threadIdx.x] = fmaxf(red[threadIdx.x], red[threadIdx.x + s]);
    __syncthreads();
  }
  if (threadIdx.x == 0) atomicMax(&bmax[b], flipf(red[0]));
}

// ---------------- pass 2: per-batch sum of exp(KQ - max), deterministic ----------------
__global__ void kq_sum_kernel(const float* __restrict__ Kp,
                              const float* __restrict__ Qp,
                              const unsigned* __restrict__ bmax,
                              float* __restrict__ partials) {
  int mg = blockIdx.x, nt = blockIdx.y, b = blockIdx.z;
  int wave = threadIdx.x >> 5, lane = threadIdx.x & 31;
  int lm = lane & 15, khh = lane >> 4;
  const float* Kb = Kp + (size_t)b * C4n * Nn;
  const float* Qb = Qp + (size_t)b * C4n * Nn;
  float bm = unflipf(bmax[b]);
  v8f t = kq_tile(Kb, Qb, nt * 16, mg * 128 + wave * 16, lm, khh);
  float s = 0.f;
#pragma unroll
  for (int r = 0; r < 8; ++r) s += __expf(t[r] - bm);
  __shared__ float red[256];
  red[threadIdx.x] = s;
  __syncthreads();
  for (int st = 128; st > 0; st >>= 1) {
    if (threadIdx.x < st) red[threadIdx.x] += red[threadIdx.x + st];
    __syncthreads();
  }
  if (threadIdx.x == 0) partials[(size_t)b * NPART + nt * MGRP + mg] = red[0];
}

__global__ void reduce_sum_kernel(const float* __restrict__ partials,
                                  float* __restrict__ bsuminv) {
  int b = blockIdx.x;
  __shared__ float red[256];
  float s = 0.f;
  for (int i = threadIdx.x; i < NPART; i += 256) s += partials[(size_t)b * NPART + i];
  red[threadIdx.x] = s;
  __syncthreads();
  for (int st = 128; st > 0; st >>= 1) {
    if (threadIdx.x < st) red[threadIdx.x] += red[threadIdx.x + st];
    __syncthreads();
  }
  if (threadIdx.x == 0) bsuminv[b] = 1.0f / red[0];
}

// ---------------- pass 3: fused exp(KQ) x V^T GEMM into output ----------------
// Block (ntile, b), 8 waves; wave w owns c-tile [16w,16w+16). Per 128-m chunk:
// wave w computes its 16-m exp subtile (WMMA, K A-frags hoisted) -> LDS, then
// all waves accumulate D[n][c] with b64-coalesced Vp fragments.
__global__ void attn_feat_kernel(const float* __restrict__ Kp,
                                 const float* __restrict__ Qp,
                                 const float* __restrict__ Vp,
                                 const unsigned* __restrict__ bmax,
                                 const float* __restrict__ bsuminv,
                                 float* __restrict__ out) {
  int nt = blockIdx.x, b = blockIdx.y;
  int wave = threadIdx.x >> 5, lane = threadIdx.x & 31;
  int lm = lane & 15, khh = lane >> 4, kh = khh << 1, rofs = khh << 3;
  int n0 = nt * 16;
  const float* Kb = Kp + (size_t)b * C4n * Nn;
  const float* Qb = Qp + (size_t)b * C4n * Nn;
  const float* Vb = Vp + (size_t)b * Nn * Cc;
  float bm = unflipf(bmax[b]);
  float inv = bsuminv[b];
  __shared__ float E[16][132];  // bank = (4n + m)%64 across rows -> spread
  v2f aK[8];
#pragma unroll
  for (int j = 0; j < 8; ++j)
    aK[j] = *(const v2f*)(Kb + (size_t)(2 * j + khh) * (2 * Nn) + 2 * (n0 + lm));
  v8f d = {};
  for (int mg = 0; mg < MGRP; ++mg) {
    int mbase = mg * 128;
    int m0 = mbase + wave * 16;
    v8f t = {};
#pragma unroll
    for (int j = 0; j < 8; ++j) {
      v2f bb = *(const v2f*)(Qb + (size_t)(2 * j + khh) * (2 * Nn) + 2 * (m0 + lm));
      t = wmma4(aK[j], bb, t);
    }
    __syncthreads();  // previous E fully consumed
#pragma unroll
    for (int r = 0; r < 8; ++r)
      E[r + rofs][wave * 16 + lm] = __expf(t[r] - bm);
    __syncthreads();  // E ready
#pragma unroll 8
    for (int mk = 0; mk < 128; mk += 4) {
      v2f a, bb;
      a.x = E[lm][mk + kh];
      a.y = E[lm][mk + kh + 1];
      bb = *(const v2f*)(Vb + (size_t)(((mbase + mk) >> 1) + khh) * (2 * Cc) +
                         2 * (wave * 16 + lm));
      d = wmma4(a, bb, d);
    }
  }
#pragma unroll
  for (int r = 0; r < 8; ++r) {
    int nn = n0 + r + rofs;
    int c = wave * 16 + lm;
    out[((size_t)b * 2 * Cc + Cc + c) * Nn + nn] = d[r] * inv;
  }
}

extern "C" void kernel_launch(void* const* d_in, const int* in_sizes, int n_in,
                              void* d_out, int out_size, void* d_ws, size_t ws_size,
                              hipStream_t stream) {
  (void)in_sizes; (void)n_in; (void)out_size; (void)ws_size;
  const float* x_en  = (const float*)d_in[0];
  const float* x_de  = (const float*)d_in[1];
  const float* x_cat = (const float*)d_in[2];
  const float* w_kq  = (const float*)d_in[3];
  const float* kq_g  = (const float*)d_in[4];
  const float* kq_b  = (const float*)d_in[5];
  const float* kq_m  = (const float*)d_in[6];
  const float* kq_v  = (const float*)d_in[7];
  const float* w_v   = (const float*)d_in[8];
  const float* v_g   = (const float*)d_in[9];
  const float* v_b   = (const float*)d_in[10];
  const float* v_m   = (const float*)d_in[11];
  const float* v_v   = (const float*)d_in[12];
  const float* w_red = (const float*)d_in[13];
  const float* r_g   = (const float*)d_in[14];
  const float* r_b   = (const float*)d_in[15];
  const float* r_m   = (const float*)d_in[16];
  const float* r_v   = (const float*)d_in[17];
  float* out = (float*)d_out;

  float* Kp = (float*)d_ws;                                 // B*C4*N (k-paired)
  float* Qp = Kp + (size_t)Bn * C4n * Nn;                   // B*C4*N (k-paired)
  float* Vp = Qp + (size_t)Bn * C4n * Nn;                   // B*N*C  (m-paired)
  unsigned* bmax = (unsigned*)(Vp + (size_t)Bn * Nn * Cc);  // B
  float* bsuminv = (float*)(bmax + Bn);                     // B
  float* partials = bsuminv + Bn;                           // B*NPART

  hipMemsetAsync(bmax, 0, Bn * sizeof(unsigned), stream);   // flip-space minimum

  conv_kq_wmma<<<dim3(NGRP, C4n / 16, 2 * Bn), 256, 0, stream>>>(
      x_en, x_de, w_kq, kq_g, kq_b, kq_m, kq_v, Kp, Qp);
  conv_red_wmma<<<dim3(NGRP, Cc / 16, Bn), 256, 0, stream>>>(
      x_cat, w_red, r_g, r_b, r_m, r_v, out);
  conv_v_wmma<<<dim3(NGRP, Cc / 16, Bn), 256, 0, stream>>>(
      out, w_v, v_g, v_b, v_m, v_v, Vp);
  kq_max_kernel<<<dim3(MGRP, NTILE, Bn), 256, 0, stream>>>(Kp, Qp, bmax);
  kq_sum_kernel<<<dim3(MGRP, NTILE, Bn), 256, 0, stream>>>(Kp, Qp, bmax, partials);
  reduce_sum_kernel<<<dim3(Bn), 256, 0, stream>>>(partials, bsuminv);
  attn_feat_kernel<<<dim3(NTILE, Bn), 256, 0, stream>>>(
      Kp, Qp, Vp, bmax, bsuminv, out);
}